// SAGE_Variant_5463198401302
// MI455X (gfx1250) — compile-verified
//
#include <hip/hip_runtime.h>

#define N_NODES 100000
#define N_EDGES 1600000
#define D 128
#define LDA 132   // 16-row LDS tile padded: 132 % 64 = 4 -> conflict-free column reads

typedef float v2f __attribute__((ext_vector_type(2)));
typedef float v8f __attribute__((ext_vector_type(8)));

__global__ void zero_f32(float* __restrict__ p, int n) {
  int i = blockIdx.x * blockDim.x + threadIdx.x;
  if (i < n) p[i] = 0.0f;
}

__global__ void degree_kernel(const long long* __restrict__ dst,
                              float* __restrict__ deg) {
  int e = blockIdx.x * blockDim.x + threadIdx.x;
  if (e < N_EDGES) atomicAdd(&deg[(int)dst[e]], 1.0f);
}

// One wave per edge: 32 lanes x float4 covers the 128-float feature row.
// Loads are fully coalesced (512B per wave), atomics are contiguous per row.
__global__ __launch_bounds__(256) void scatter_kernel(
    const float* __restrict__ h,
    const long long* __restrict__ src,
    const long long* __restrict__ dst,
    float* __restrict__ agg) {
  int tid = blockIdx.x * blockDim.x + threadIdx.x;   // N_EDGES * 32 threads
  int e  = tid >> 5;
  int d0 = (tid & 31) * 4;
  int s = (int)src[e];
  int t = (int)dst[e];
  const float4 v = *(const float4*)(h + (size_t)s * D + d0);
  float* ap = agg + (size_t)t * D + d0;
  atomicAdd(ap + 0, v.x);
  atomicAdd(ap + 1, v.y);
  atomicAdd(ap + 2, v.z);
  atomicAdd(ap + 3, v.w);
}

// out = act( (agg/max(deg,1)) @ Wl + bl + h @ Wr )
// Block = 8 waves; block owns 16 output rows, wave w owns cols [16w,16w+16).
// fp32 WMMA 16x16x4: A fragment = 2 VGPRs (M=lane&15, K pair selected by lane>=16),
// B fragment = 2 VGPRs (N=lane&15, same K-pair convention). 64 WMMA per wave.
template <int RELU>
__global__ __launch_bounds__(256) void sage_gemm(
    const float* __restrict__ h,    // [N,D] layer input (lin_r operand)
    const float* __restrict__ agg,  // [N,D] neighbor sum
    const float* __restrict__ deg,  // [N]
    const float* __restrict__ Wl,   // [D,D]
    const float* __restrict__ bl,   // [D]
    const float* __restrict__ Wr,   // [D,D]
    float* __restrict__ out) {      // [N,D]
  __shared__ float sAgg[16 * LDA];
  __shared__ float sH[16 * LDA];

  const int row0 = blockIdx.x * 16;
  const int t = threadIdx.x;

  // Stage both 16x128 activation tiles; fold 1/max(deg,1) into the agg tile.
  for (int i = t; i < 16 * D; i += 256) {
    int r = i >> 7;
    int cc = i & 127;
    float id = 1.0f / fmaxf(deg[row0 + r], 1.0f);
    sAgg[r * LDA + cc] = agg[(size_t)(row0 + r) * D + cc] * id;
    sH[r * LDA + cc]   = h[(size_t)(row0 + r) * D + cc];
  }
  __syncthreads();

  const int wave = t >> 5;          // 0..7 -> column tile
  const int lane = t & 31;
  const int half = lane >> 4;       // which K pair (0: K,K+1; 1: K+2,K+3)
  const int lp   = lane & 15;       // M index (A) / N index (B)
  const int col0 = wave * 16;

  v8f acc = {};
  const float* WlCol = Wl + col0 + lp;   // column lp of this tile
  const float* WrCol = Wr + col0 + lp;

#pragma unroll
  for (int k = 0; k < D; k += 4) {
    const int kk = k + 2 * half;
    v2f a, b;
    a.x = sAgg[lp * LDA + kk];
    a.y = sAgg[lp * LDA + kk + 1];
    b.x = WlCol[(size_t)kk * D];
    b.y = WlCol[(size_t)(kk + 1) * D];
    acc = __builtin_amdgcn_wmma_f32_16x16x4_f32(false, a, false, b,
                                                (short)0, acc, false, false);
  }
#pragma unroll
  for (int k = 0; k < D; k += 4) {
    const int kk = k + 2 * half;
    v2f a, b;
    a.x = sH[lp * LDA + kk];
    a.y = sH[lp * LDA + kk + 1];
    b.x = WrCol[(size_t)kk * D];
    b.y = WrCol[(size_t)(kk + 1) * D];
    acc = __builtin_amdgcn_wmma_f32_16x16x4_f32(false, a, false, b,
                                                (short)0, acc, false, false);
  }

  const float bias = bl[col0 + lp];
  // C/D layout: VGPR v -> row (v + 8*half), N = lane&15
#pragma unroll
  for (int v = 0; v < 8; v++) {
    float val = acc[v] + bias;
    if (RELU) val = fmaxf(val, 0.0f);
    out[(size_t)(row0 + v + 8 * half) * D + col0 + lp] = val;
  }
}

extern "C" void kernel_launch(void* const* d_in, const int* in_sizes, int n_in,
                              void* d_out, int out_size, void* d_ws, size_t ws_size,
                              hipStream_t stream) {
  (void)in_sizes; (void)n_in; (void)out_size; (void)ws_size;

  const float* x = (const float*)d_in[0];
  const long long* ei = (const long long*)d_in[1];   // int64 [2, E]
  const long long* src = ei;
  const long long* dst = ei + N_EDGES;
  const float* Wl[3] = {(const float*)d_in[2], (const float*)d_in[5], (const float*)d_in[8]};
  const float* bl[3] = {(const float*)d_in[3], (const float*)d_in[6], (const float*)d_in[9]};
  const float* Wr[3] = {(const float*)d_in[4], (const float*)d_in[7], (const float*)d_in[10]};

  float* deg = (float*)d_ws;
  float* agg = deg + N_NODES;
  float* h1  = agg + (size_t)N_NODES * D;
  float* h2  = h1 + (size_t)N_NODES * D;
  float* out = (float*)d_out;

  // Degree once (edge structure fixed across layers)
  zero_f32<<<(N_NODES + 255) / 256, 256, 0, stream>>>(deg, N_NODES);
  degree_kernel<<<N_EDGES / 256, 256, 0, stream>>>(dst, deg);

  const float* hin = x;
  float* houts[3] = {h1, h2, out};
  for (int l = 0; l < 3; l++) {
    zero_f32<<<(N_NODES * D) / 256, 256, 0, stream>>>(agg, N_NODES * D);
    scatter_kernel<<<(N_EDGES / 256) * 32, 256, 0, stream>>>(hin, src, dst, agg);
    if (l < 2)
      sage_gemm<1><<<N_NODES / 16, 256, 0, stream>>>(hin, agg, deg, Wl[l], bl[l], Wr[l], houts[l]);
    else
      sage_gemm<0><<<N_NODES / 16, 256, 0, stream>>>(hin, agg, deg, Wl[l], bl[l], Wr[l], houts[l]);
    hin = houts[l];
  }
}